// SCSSystem_57105885167627
// MI455X (gfx1250) — compile-verified
//
#include <hip/hip_runtime.h>

#define Bn 8
#define Hn 256
#define Wn 256
#define Pn 8
#define PHn 32
#define Nn 1024
#define Sn 64
#define Tn 64

typedef __attribute__((ext_vector_type(2))) float v2f;
typedef __attribute__((ext_vector_type(8))) float v8f;

__device__ __forceinline__ float wave_sum32(float v) {
    v += __shfl_xor(v, 16, 32);
    v += __shfl_xor(v, 8, 32);
    v += __shfl_xor(v, 4, 32);
    v += __shfl_xor(v, 2, 32);
    v += __shfl_xor(v, 1, 32);
    return v;
}

__device__ __forceinline__ float wave_max32(float v) {
    v = fmaxf(v, __shfl_xor(v, 16, 32));
    v = fmaxf(v, __shfl_xor(v, 8, 32));
    v = fmaxf(v, __shfl_xor(v, 4, 32));
    v = fmaxf(v, __shfl_xor(v, 2, 32));
    v = fmaxf(v, __shfl_xor(v, 1, 32));
    return v;
}

__global__ __launch_bounds__(256) void scs_patch_kernel(
    const float* __restrict__ spikes,   // [B,H,W]
    const float* __restrict__ Wdyn,     // [B,N,T,S]
    const float* __restrict__ ln_gamma, // [T]
    const float* __restrict__ ln_beta,  // [T]
    const float* __restrict__ gates,    // [N]
    const float* __restrict__ biases,   // [N]
    float* __restrict__ out)            // [B,H,W]
{
    __shared__ float lds_x[8][64];
    __shared__ float lds_o[8][64];

    const int tid   = threadIdx.x;
    const int wave  = tid >> 5;
    const int lane  = tid & 31;
    const int patch = (blockIdx.x << 3) + wave;   // 0..8191
    const int b     = patch >> 10;
    const int n     = patch & (Nn - 1);
    const int py    = n >> 5;
    const int px    = n & 31;

    // ---------- load 8x8 patch (row-major pixels, matches unfold) ----------
    const int baseHW = (b * Hn + py * Pn) * Wn + px * Pn;
    const int sA = lane;        // pixel 0..31
    const int sB = lane + 32;   // pixel 32..63
    const float xA = spikes[baseHW + (sA >> 3) * Wn + (sA & 7)];
    const float xB = spikes[baseHW + (sB >> 3) * Wn + (sB & 7)];
    lds_x[wave][sA] = xA;
    lds_x[wave][sB] = xB;
    const float psum = wave_sum32(xA + xB);       // sum of all 64 pixels
    asm volatile("s_wait_dscnt 0" ::: "memory");  // LDS writes visible within wave

    // ---------- 64x64 mat-vec via V_WMMA_F32_16X16X4_F32 ----------
    // A layout (16x4 f32): lanes 0-15 rows M, v0=K0 v1=K1; lanes 16-31: v0=K2 v1=K3
    // B layout (4x16 f32): lanes 0-15: v0=K0 v1=K1 (N=lane); lanes 16-31: v0=K2 v1=K3
    const float* Wp  = Wdyn + (size_t)patch * (size_t)(Tn * Sn);
    const int m   = lane & 15;
    const int hi2 = (lane >> 4) << 1;             // 0 or 2
    const float* w0 = Wp + (m +  0) * Sn + hi2;
    const float* w1 = Wp + (m + 16) * Sn + hi2;
    const float* w2 = Wp + (m + 32) * Sn + hi2;
    const float* w3 = Wp + (m + 48) * Sn + hi2;
    const float* xp = &lds_x[wave][hi2];

    v8f c0 = {0.f,0.f,0.f,0.f,0.f,0.f,0.f,0.f};
    v8f c1 = c0, c2 = c0, c3 = c0;

    #pragma unroll 4
    for (int kc = 0; kc < 16; ++kc) {
        const int s0 = kc << 2;
        const v2f bv = *(const v2f*)(xp + s0);    // ds_load_b64: x[s0+hi2..+1]
        const v2f a0 = *(const v2f*)(w0 + s0);    // global_load_b64 each
        const v2f a1 = *(const v2f*)(w1 + s0);
        const v2f a2 = *(const v2f*)(w2 + s0);
        const v2f a3 = *(const v2f*)(w3 + s0);
        c0 = __builtin_amdgcn_wmma_f32_16x16x4_f32(false, a0, false, bv, (short)0, c0, false, false);
        c1 = __builtin_amdgcn_wmma_f32_16x16x4_f32(false, a1, false, bv, (short)0, c1, false, false);
        c2 = __builtin_amdgcn_wmma_f32_16x16x4_f32(false, a2, false, bv, (short)0, c2, false, false);
        c3 = __builtin_amdgcn_wmma_f32_16x16x4_f32(false, a3, false, bv, (short)0, c3, false, false);
    }

    // ---------- scatter C tiles to LDS (columns of C are identical) ----------
    // C layout: VGPR r, lanes 0-15 -> M=r, lanes 16-31 -> M=r+8
    const int hiOff = (lane >> 4) << 3;           // 0 or 8
    #pragma unroll
    for (int r = 0; r < 8; ++r) {
        if (m == r) {
            lds_o[wave][ 0 + hiOff + r] = c0[r];
            lds_o[wave][16 + hiOff + r] = c1[r];
            lds_o[wave][32 + hiOff + r] = c2[r];
            lds_o[wave][48 + hiOff + r] = c3[r];
        }
    }
    asm volatile("s_wait_dscnt 0" ::: "memory");

    const float oA = lds_o[wave][sA];
    const float oB = lds_o[wave][sB];

    // ---------- LayerNorm over T=64 ----------
    const float mu   = wave_sum32(oA + oB) * (1.0f / 64.0f);
    const float dA   = oA - mu;
    const float dB   = oB - mu;
    const float var  = wave_sum32(dA * dA + dB * dB) * (1.0f / 64.0f);
    const float rstd = rsqrtf(var + 1e-5f);
    const float nA   = dA * rstd * ln_gamma[sA] + ln_beta[sA];
    const float nB   = dB * rstd * ln_gamma[sB] + ln_beta[sB];

    // ---------- temperature softmax (1/TEMP = 10) ----------
    const float zA  = nA * 10.0f;
    const float zB  = nB * 10.0f;
    const float mx  = wave_max32(fmaxf(zA, zB));
    const float eA  = __expf(zA - mx);
    const float eB  = __expf(zB - mx);
    const float inv = 1.0f / wave_sum32(eA + eB);

    // ---------- gated affine + fold back ----------
    const float affine = gates[n] * psum + biases[n];
    out[baseHW + (sA >> 3) * Wn + (sA & 7)] = eA * inv * affine;
    out[baseHW + (sB >> 3) * Wn + (sB & 7)] = eB * inv * affine;
}

extern "C" void kernel_launch(void* const* d_in, const int* in_sizes, int n_in,
                              void* d_out, int out_size, void* d_ws, size_t ws_size,
                              hipStream_t stream) {
    const float* spikes   = (const float*)d_in[0];
    const float* Wdyn     = (const float*)d_in[1];
    const float* ln_gamma = (const float*)d_in[2];
    const float* ln_beta  = (const float*)d_in[3];
    const float* gates    = (const float*)d_in[4];
    const float* biases   = (const float*)d_in[5];
    float* out            = (float*)d_out;

    // 8192 patches, 1 wave32 per patch, 8 waves per block
    scs_patch_kernel<<<dim3(1024), dim3(256), 0, stream>>>(
        spikes, Wdyn, ln_gamma, ln_beta, gates, biases, out);
}